// PolylineSubgraphEncoder_670014898400
// MI455X (gfx1250) — compile-verified
//
#include <hip/hip_runtime.h>
#include <math.h>

typedef float v2f __attribute__((ext_vector_type(2)));
typedef float v8f __attribute__((ext_vector_type(8)));

#define DH 128   // hidden width (both layers' output width)

// ---------------------------------------------------------------------------
// Degree / normalization kernels
// ---------------------------------------------------------------------------
__global__ void init_deg_kernel(float* __restrict__ deg, int n) {
    int i = blockIdx.x * blockDim.x + threadIdx.x;
    if (i < n) deg[i] = 1.0f;                // self-loop contributes 1
}

__global__ void deg_accum_kernel(const int* __restrict__ dst, float* __restrict__ deg, int e) {
    int i = blockIdx.x * blockDim.x + threadIdx.x;
    if (i < e) atomicAdd(&deg[dst[i]], 1.0f);
}

__global__ void dinv_kernel(float* __restrict__ deg, int n) {
    int i = blockIdx.x * blockDim.x + threadIdx.x;
    if (i < n) deg[i] = rsqrtf(deg[i]);      // in place: deg -> deg^{-1/2}
}

// ---------------------------------------------------------------------------
// fp32 WMMA GEMM: C[n_rows x DH] = A[n_rows x K] * W[K x DH]
// One wave per 16-row strip: 8 accumulator tiles (16x128 strip), so A is
// loaded ONCE per K-step and reused by 8 independent WMMAs (no RAW hazards,
// 8x less A traffic than tile-per-wave).
// ---------------------------------------------------------------------------
template <int K>
__global__ void gemm_wmma_f32_kernel(const float* __restrict__ A,
                                     const float* __restrict__ W,
                                     float* __restrict__ C,
                                     int n_rows) {
    int wave = blockIdx.x * (blockDim.x >> 5) + (threadIdx.x >> 5);
    int tiles_m = (n_rows + 15) >> 4;
    if (wave >= tiles_m) return;                       // wave-uniform: EXEC stays all-1s

    int lane  = threadIdx.x & 31;
    int lhalf = lane >> 4;                             // 0: K pair {0,1}; 1: K pair {2,3}
    int lidx  = lane & 15;

    // A 16x4 layout: lanes 0-15 hold K={k,k+1}, lanes 16-31 hold K={k+2,k+3}
    const float* arow = A + (size_t)(wave * 16 + lidx) * K + 2 * lhalf;

    v8f acc[8] = {};                                   // column tiles tn=0..7
    for (int k = 0; k < K; k += 4) {
        v2f a;
        a[0] = arow[k + 0];
        a[1] = arow[k + 1];
        // B 4x16 layout: vgpr0 = rows k/k+1 striped over lane halves, vgpr1 = rows k+2/k+3
        const float* wr0 = W + (size_t)(k + lhalf) * DH + lidx;
        const float* wr1 = W + (size_t)(k + 2 + lhalf) * DH + lidx;
        #pragma unroll
        for (int tn = 0; tn < 8; ++tn) {
            v2f b;
            b[0] = wr0[tn * 16];
            b[1] = wr1[tn * 16];
            acc[tn] = __builtin_amdgcn_wmma_f32_16x16x4_f32(
                /*neg_a=*/false, a, /*neg_b=*/false, b,
                /*c_mod=*/(short)0, acc[tn], /*reuse_a=*/false, /*reuse_b=*/false);
        }
    }

    // C/D layout: VGPR r, lanes 0-15 -> M=r, lanes 16-31 -> M=r+8
    int row0 = wave * 16 + lhalf * 8;
    #pragma unroll
    for (int tn = 0; tn < 8; ++tn) {
        #pragma unroll
        for (int r = 0; r < 8; ++r)
            C[(size_t)(row0 + r) * DH + tn * 16 + lidx] = acc[tn][r];
    }
}

// ---------------------------------------------------------------------------
// Aggregation kernels
// ---------------------------------------------------------------------------
// Seed accumulator with self-loop term: agg[i] = xw[i] * dinv[i]^2
__global__ void self_init_kernel(const float* __restrict__ xw,
                                 const float* __restrict__ dinv,
                                 float* __restrict__ agg, int n) {
    int t = blockIdx.x * blockDim.x + threadIdx.x;     // one float4 per thread
    int node = t >> 5;                                 // 32 float4s per row (DH=128)
    if (node >= n) return;
    int q = t & 31;
    float di = dinv[node];
    float s = di * di;
    float4 v = ((const float4*)(xw + (size_t)node * DH))[q];
    v.x *= s; v.y *= s; v.z *= s; v.w *= s;
    ((float4*)(agg + (size_t)node * DH))[q] = v;
}

// One wave per edge: coalesced float4 gather of xw[src], scaled atomic scatter to agg[dst]
__global__ void edge_scatter_kernel(const int* __restrict__ src,
                                    const int* __restrict__ dst,
                                    const float* __restrict__ dinv,
                                    const float* __restrict__ xw,
                                    float* __restrict__ agg, int e) {
    int wid = blockIdx.x * (blockDim.x >> 5) + (threadIdx.x >> 5);
    if (wid >= e) return;
    int lane = threadIdx.x & 31;
    int s = src[wid];
    int d = dst[wid];
    float norm = dinv[s] * dinv[d];
    float4 m = ((const float4*)(xw + (size_t)s * DH))[lane];
    float* out = agg + (size_t)d * DH + lane * 4;
    atomicAdd(out + 0, m.x * norm);
    atomicAdd(out + 1, m.y * norm);
    atomicAdd(out + 2, m.z * norm);
    atomicAdd(out + 3, m.w * norm);
}

__global__ void bias_relu_kernel(float* __restrict__ agg,
                                 const float* __restrict__ b, int total) {
    int i = blockIdx.x * blockDim.x + threadIdx.x;
    if (i >= total) return;
    float v = agg[i] + b[i & (DH - 1)];
    agg[i] = fmaxf(v, 0.0f);
}

// ---------------------------------------------------------------------------
// Launch
// ---------------------------------------------------------------------------
extern "C" void kernel_launch(void* const* d_in, const int* in_sizes, int n_in,
                              void* d_out, int out_size, void* d_ws, size_t ws_size,
                              hipStream_t stream) {
    const float* x   = (const float*)d_in[0];
    const int*   ei  = (const int*)d_in[1];   // [2, E] int32
    const float* W1  = (const float*)d_in[2];
    const float* b1  = (const float*)d_in[3];
    const float* W2  = (const float*)d_in[4];
    const float* b2  = (const float*)d_in[5];
    float* out = (float*)d_out;

    const int DIN = 96;
    const int N = in_sizes[0] / DIN;          // 50000
    const int E = in_sizes[1] / 2;            // 800000
    const int* src = ei;
    const int* dst = ei + E;

    // Workspace layout (256B-aligned slices)
    char* ws = (char*)d_ws;
    size_t off = 0;
    float* dinv = (float*)(ws + off); off += ((size_t)N * 4 + 255) & ~(size_t)255;
    float* buf0 = (float*)(ws + off); off += (size_t)N * DH * 4;   // xw
    float* buf1 = (float*)(ws + off);                              // agg / h

    const int T = 256;
    // --- normalization ---
    init_deg_kernel<<<(N + T - 1) / T, T, 0, stream>>>(dinv, N);
    deg_accum_kernel<<<(E + T - 1) / T, T, 0, stream>>>(dst, dinv, E);
    dinv_kernel<<<(N + T - 1) / T, T, 0, stream>>>(dinv, N);

    int tiles_m = (N + 15) / 16;                       // one wave per 16-row strip
    int gblk    = (tiles_m + 3) / 4;                   // 4 waves (128 thr) per block
    int eblk    = (E + 7) / 8;                         // 8 edge-waves per block
    int nq      = N * (DH / 4);                        // float4 elements
    int ntot    = N * DH;

    // --- layer 1 ---
    gemm_wmma_f32_kernel<96><<<gblk, 128, 0, stream>>>(x, W1, buf0, N);
    self_init_kernel<<<(nq + T - 1) / T, T, 0, stream>>>(buf0, dinv, buf1, N);
    edge_scatter_kernel<<<eblk, 256, 0, stream>>>(src, dst, dinv, buf0, buf1, E);
    bias_relu_kernel<<<(ntot + T - 1) / T, T, 0, stream>>>(buf1, b1, ntot);

    // --- layer 2 (aggregate straight into d_out) ---
    gemm_wmma_f32_kernel<128><<<gblk, 128, 0, stream>>>(buf1, W2, buf0, N);
    self_init_kernel<<<(nq + T - 1) / T, T, 0, stream>>>(buf0, dinv, out, N);
    edge_scatter_kernel<<<eblk, 256, 0, stream>>>(src, dst, dinv, buf0, out, E);
    bias_relu_kernel<<<(ntot + T - 1) / T, T, 0, stream>>>(out, b2, ntot);
}